// MulSupCosineLossCustom_87256555585551
// MI455X (gfx1250) — compile-verified
//
#include <hip/hip_runtime.h>
#include <math.h>

#define BSZ      2048
#define NV       2
#define DIM      512
#define NROW     4096         // NV * BSZ
#define NLAB     19
#define NLAB_PAD 20
#define TEMP     0.07f
#define LSCALE   1.0f         // TEMPERATURE / BASE_TEMPERATURE

typedef float v2f __attribute__((ext_vector_type(2)));
typedef float v8f __attribute__((ext_vector_type(8)));

static __device__ __forceinline__ int memrow_of(int r) {
  // contrast row r = v*BSZ + i  -> features flat row (i*NV + v)
  int v = r >> 11;          // r / 2048
  int i = r & 2047;         // r % 2048
  return i * NV + v;
}

// ---------------------------------------------------------------- kernel 1
__global__ void prep_labels_kernel(const int* __restrict__ labels,
                                   float* __restrict__ nlab) {
  int i = blockIdx.x * blockDim.x + threadIdx.x;
  if (i >= BSZ) return;
  float vals[NLAB];
  float ss = 0.f;
  #pragma unroll
  for (int k = 0; k < NLAB; ++k) {
    float v = (float)labels[i * NLAB + k];
    vals[k] = v;
    ss += v * v;
  }
  float inv = (ss > 0.f) ? (1.0f / sqrtf(ss)) : 0.f;
  #pragma unroll
  for (int k = 0; k < NLAB; ++k) nlab[i * NLAB_PAD + k] = vals[k] * inv;
  nlab[i * NLAB_PAD + NLAB] = 0.f;  // pad slot
}

// ---------------------------------------------------------------- kernel 2
__global__ void build_mask_kernel(const float* __restrict__ nlab,
                                  float* __restrict__ mask) {
  unsigned idx = blockIdx.x * blockDim.x + threadIdx.x;   // < 2048*2048
  int i = (int)(idx >> 11);
  int j = (int)(idx & 2047);
  const float* a = nlab + i * NLAB_PAD;
  const float* b = nlab + j * NLAB_PAD;
  float s = 0.f;
  #pragma unroll
  for (int k = 0; k < NLAB; ++k) s += a[k] * b[k];
  mask[idx] = s;
}

// ---------------------------------------------------------------- kernel 3
#define KCHUNK 16
#define NTILE  4     // column tiles (16 cols each) per wave, shared A fragment
#define AST    516   // A block LDS row stride (floats): 516 % 64 == 4 -> conflict-free
#define BST    18    // B chunk LDS row stride:           18*r mod 64 distinct, r<16
#define TST    17    // C tile LDS row stride

__global__ __launch_bounds__(128)
void supcon_main_kernel(const float* __restrict__ feat,
                        const float* __restrict__ mask,
                        float* __restrict__ rowloss) {
  __shared__ float Ablk[16 * AST];                 // 33,024 B
  __shared__ float Bbuf[4][NTILE * 16 * BST];      // 18,432 B
  __shared__ float Ctile[4][16 * TST];             //  4,352 B
  __shared__ float Comb[4][16][4];                 //  1,024 B

  const int tid  = threadIdx.x;
  const int wave = tid >> 5;
  const int lane = tid & 31;
  const int lm   = lane & 15;   // row (A) / col (B) within fragment
  const int hi   = lane >> 4;   // K half-select
  const int i0   = blockIdx.x * 16;

  // ---- stage anchor A block: 16 rows x 512 f32 (coalesced float4) ----
  for (int idx = tid; idx < 16 * (DIM / 4); idx += 128) {
    int row = idx >> 7;       // / (DIM/4)
    int c4  = idx & 127;
    const float4 v =
        ((const float4*)(feat + (size_t)memrow_of(i0 + row) * DIM))[c4];
    float* dst = &Ablk[row * AST + c4 * 4];
    dst[0] = v.x; dst[1] = v.y; dst[2] = v.z; dst[3] = v.w;
  }
  __syncthreads();

  const float invT  = 1.0f / TEMP;
  const int   myrow = i0 + lm;
  const int   rmod  = myrow & 2047;
  const float* maskrow = mask + (size_t)rmod * BSZ;

  // online per-row state (owned by lanes 0..15)
  float m = -3.402823466e38f, Z = 0.f, sml = 0.f, sm = 0.f;

  // each wave covers 64 columns per group; 4 waves -> 256 cols/group; 16 groups
  for (int g = 0; g < NROW / (4 * NTILE * 16); ++g) {
    const int jg0 = (g * 4 + wave) * (NTILE * 16);   // wave's 64-col strip base
    v8f c[NTILE];
    #pragma unroll
    for (int t = 0; t < NTILE; ++t) c[t] = (v8f){};

    for (int kc = 0; kc < DIM; kc += KCHUNK) {
      // stage B chunk: 64 rows (jg0..jg0+63) x 16 K-cols (coalesced float4)
      #pragma unroll
      for (int q = lane; q < NTILE * 16 * (KCHUNK / 4); q += 32) {
        int row = q >> 2;     // / (KCHUNK/4)
        int c4  = q & 3;
        const float4 v =
            ((const float4*)(feat + (size_t)memrow_of(jg0 + row) * DIM + kc))[c4];
        float* dst = &Bbuf[wave][row * BST + c4 * 4];
        dst[0] = v.x; dst[1] = v.y; dst[2] = v.z; dst[3] = v.w;
      }
      __syncthreads();
      #pragma unroll
      for (int kk = 0; kk < KCHUNK; kk += 4) {
        const int ko = kk + 2 * hi;
        v2f a = *(const v2f*)&Ablk[lm * AST + kc + ko];   // one A per 4 WMMAs
        #pragma unroll
        for (int t = 0; t < NTILE; ++t) {
          v2f b = *(const v2f*)&Bbuf[wave][(t * 16 + lm) * BST + ko];
          c[t] = __builtin_amdgcn_wmma_f32_16x16x4_f32(
              /*neg_a=*/false, a, /*neg_b=*/false, b,
              /*c_mod=*/(short)0, c[t], /*reuse_a=*/false, /*reuse_b=*/false);
        }
      }
      __syncthreads();
    }

    // spill + process the 4 tiles (fully unrolled -> static c[t] indexing,
    // no v_movrel; barriers remain uniform across all waves)
    #pragma unroll
    for (int t = 0; t < NTILE; ++t) {
      const int j0 = jg0 + t * 16;
      #pragma unroll
      for (int v = 0; v < 8; ++v)
        Ctile[wave][(v + 8 * hi) * TST + lm] = c[t][v];
      __syncthreads();
      if (lane < 16) {
        #pragma unroll 4
        for (int cc = 0; cc < 16; ++cc) {
          int   j    = j0 + cc;
          float l    = Ctile[wave][lm * TST + cc] * invT;
          bool  diag = (j == myrow);
          float nm = fmaxf(m, l);                 // max INCLUDES diagonal
          float e  = diag ? 0.f : __expf(l - nm); // Z excludes diagonal
          Z  = Z * __expf(m - nm) + e;
          m  = nm;
          float mk = diag ? 0.f : maskrow[j & 2047];
          sml = fmaf(mk, l, sml);
          sm += mk;
        }
      }
      __syncthreads();
    }
  }

  // ---- combine the 4 waves' partials per row ----
  if (lane < 16) {
    Comb[wave][lm][0] = m;   Comb[wave][lm][1] = Z;
    Comb[wave][lm][2] = sml; Comb[wave][lm][3] = sm;
  }
  __syncthreads();
  if (tid < 16) {
    float gm = -3.402823466e38f;
    #pragma unroll
    for (int w = 0; w < 4; ++w) gm = fmaxf(gm, Comb[w][tid][0]);
    float gZ = 0.f, gsml = 0.f, gsm = 0.f;
    #pragma unroll
    for (int w = 0; w < 4; ++w) {
      gZ   += Comb[w][tid][1] * __expf(Comb[w][tid][0] - gm);
      gsml += Comb[w][tid][2];
      gsm  += Comb[w][tid][3];
    }
    float logZ  = __logf(gZ);
    float numer = gsml - (gm + logZ) * gsm;   // sum mask*(l - m - logZ)
    rowloss[i0 + tid] = -LSCALE * (numer / gsm);
  }
}

// ---------------------------------------------------------------- kernel 4
__global__ void finalize_kernel(const float* __restrict__ rowloss,
                                float* __restrict__ out) {
  __shared__ float red[256];
  int tid = threadIdx.x;
  float s = 0.f;
  for (int i = tid; i < NROW; i += 256) s += rowloss[i];
  red[tid] = s;
  __syncthreads();
  for (int off = 128; off > 0; off >>= 1) {
    if (tid < off) red[tid] += red[tid + off];
    __syncthreads();
  }
  if (tid == 0) out[0] = red[0] / (float)NROW;
}

// ---------------------------------------------------------------- launch
extern "C" void kernel_launch(void* const* d_in, const int* in_sizes, int n_in,
                              void* d_out, int out_size, void* d_ws, size_t ws_size,
                              hipStream_t stream) {
  (void)in_sizes; (void)n_in; (void)out_size; (void)ws_size;
  const float* feat   = (const float*)d_in[0];   // (2048, 2, 512) f32
  const int*   labels = (const int*)d_in[1];     // (2048, 19) i32
  float*       out    = (float*)d_out;           // scalar f32

  char* ws = (char*)d_ws;
  float* nlab    = (float*)(ws);                                   // 2048*20 f32
  float* mask    = (float*)(ws + (size_t)BSZ * NLAB_PAD * 4);      // 2048*2048 f32
  float* rowloss = (float*)(ws + (size_t)BSZ * NLAB_PAD * 4
                               + (size_t)BSZ * BSZ * 4);           // 4096 f32

  prep_labels_kernel<<<(BSZ + 255) / 256, 256, 0, stream>>>(labels, nlab);
  build_mask_kernel<<<(BSZ * BSZ) / 256, 256, 0, stream>>>(nlab, mask);
  supcon_main_kernel<<<NROW / 16, 128, 0, stream>>>(feat, mask, rowloss);
  finalize_kernel<<<1, 256, 0, stream>>>(rowloss, out);
}